// KalmanFilter_28991029248442
// MI455X (gfx1250) — compile-verified
//
#include <hip/hip_runtime.h>

// Kalman filter forward pass on gfx1250 (MI455X).
// 1 workgroup (4 wave32) per batch element; all per-step 32x32 matmuls via
// V_WMMA_F32_16X16X4_F32 (f32 precision), one 16x16 output tile per wave.
// Matrices resident in LDS with row stride 34 (bank-conflict-free + b64-aligned).
// Fragments are preloaded in bulk so the 8-deep WMMA accumulation chain issues
// back-to-back instead of stalling on LDS latency per instruction.

typedef __attribute__((ext_vector_type(2))) float v2f;
typedef __attribute__((ext_vector_type(8))) float v8f;

constexpr int Nn = 32;   // state dim
constexpr int Mm = 16;   // control dim
constexpr int Pp = 32;   // obs dim
constexpr int Tt = 200;  // timesteps
constexpr int Bb = 256;  // batch
constexpr int LD = 34;   // LDS row stride (floats): distinct banks per row-lane, 8B-aligned rows

// ---- WMMA fragment helpers (16x16x4 f32 tiles from LDS, row-major stride LD) ----

// A-matrix 16x4: lane<16 holds row=row0+lane cols {k0,k0+1}; lane>=16 holds cols {k0+2,k0+3}
__device__ __forceinline__ v2f frag_a(const float* m, int row0, int k0) {
  int lane = threadIdx.x & 31;
  int r = row0 + (lane & 15);
  int k = k0 + ((lane >> 4) << 1);
  v2f a;
  a.x = m[r * LD + k];
  a.y = m[r * LD + k + 1];
  return a;
}
// B-matrix 4x16: VGPR0 = rows {k0 (lanes0-15), k0+2 (lanes16-31)}, VGPR1 = {k0+1, k0+3}
__device__ __forceinline__ v2f frag_b(const float* m, int k0, int col0) {
  int lane = threadIdx.x & 31;
  int c = col0 + (lane & 15);
  int k = k0 + ((lane >> 4) << 1);
  v2f b;
  b.x = m[k * LD + c];
  b.y = m[(k + 1) * LD + c];
  return b;
}
// B-fragment of M^T (i.e. right-multiply by M^T): B[k][n] = M[n][k]
__device__ __forceinline__ v2f frag_bt(const float* m, int k0, int col0) {
  int lane = threadIdx.x & 31;
  int c = col0 + (lane & 15);
  int k = k0 + ((lane >> 4) << 1);
  v2f b;
  b.x = m[c * LD + k];
  b.y = m[c * LD + k + 1];
  return b;
}
// C/D 16x16: VGPR v = rows {row0+v (lanes0-15), row0+v+8 (lanes16-31)}, col = col0+lane%16
__device__ __forceinline__ v8f frag_c(const float* m, int row0, int col0) {
  int lane = threadIdx.x & 31;
  int c = col0 + (lane & 15);
  int rbase = row0 + ((lane >> 4) << 3);
  v8f acc;
#pragma unroll
  for (int v = 0; v < 8; ++v) acc[v] = m[(rbase + v) * LD + c];
  return acc;
}
__device__ __forceinline__ void store_c(float* m, int row0, int col0, v8f acc) {
  int lane = threadIdx.x & 31;
  int c = col0 + (lane & 15);
  int rbase = row0 + ((lane >> 4) << 3);
#pragma unroll
  for (int v = 0; v < 8; ++v) m[(rbase + v) * LD + c] = acc[v];
}

// D = A * (bT ? B^T : B) [+ Cadd], all 32x32 in LDS. Each wave computes its
// own 16x16 tile. All 8 fragment pairs are loaded up-front so the 8 chained
// WMMAs issue without per-instruction LDS waits.
__device__ __forceinline__ void mm32(float* D, const float* A, const float* B,
                                     const float* Cadd, bool bT) {
  int wv = threadIdx.x >> 5;
  int row0 = (wv >> 1) << 4;
  int col0 = (wv & 1) << 4;
  v2f af[8], bf[8];
#pragma unroll
  for (int i = 0; i < 8; ++i) {
    af[i] = frag_a(A, row0, 4 * i);
    bf[i] = bT ? frag_bt(B, 4 * i, col0) : frag_b(B, 4 * i, col0);
  }
  v8f acc;
  if (Cadd) {
    acc = frag_c(Cadd, row0, col0);
  } else {
#pragma unroll
    for (int v = 0; v < 8; ++v) acc[v] = 0.f;
  }
#pragma unroll
  for (int i = 0; i < 8; ++i) {
    acc = __builtin_amdgcn_wmma_f32_16x16x4_f32(false, af[i], false, bf[i],
                                                (short)0, acc, false, false);
  }
  store_c(D, row0, col0, acc);
}

// D = A1 * B1^T + A2 * B2^T (fused accumulator: Joseph-form covariance update)
__device__ __forceinline__ void mm32_two_bt(float* D, const float* A1, const float* B1,
                                            const float* A2, const float* B2) {
  int wv = threadIdx.x >> 5;
  int row0 = (wv >> 1) << 4;
  int col0 = (wv & 1) << 4;
  v2f af[16], bf[16];
#pragma unroll
  for (int i = 0; i < 8; ++i) {
    af[i] = frag_a(A1, row0, 4 * i);
    bf[i] = frag_bt(B1, 4 * i, col0);
    af[8 + i] = frag_a(A2, row0, 4 * i);
    bf[8 + i] = frag_bt(B2, 4 * i, col0);
  }
  v8f acc;
#pragma unroll
  for (int v = 0; v < 8; ++v) acc[v] = 0.f;
#pragma unroll
  for (int i = 0; i < 16; ++i) {
    acc = __builtin_amdgcn_wmma_f32_16x16x4_f32(false, af[i], false, bf[i],
                                                (short)0, acc, false, false);
  }
  store_c(D, row0, col0, acc);
}

// In-place symmetrize 0.5*(M + M^T); each (i<=j) pair handled by exactly one thread.
__device__ __forceinline__ void symmetrize(float* M) {
  for (int e = threadIdx.x; e < 1024; e += 128) {
    int i = e >> 5, j = e & 31;
    if (i <= j) {
      float a = M[i * LD + j];
      float b = M[j * LD + i];
      float v = 0.5f * (a + b);
      M[i * LD + j] = v;
      M[j * LD + i] = v;
    }
  }
}

__global__ __launch_bounds__(128) void kalman_fwd_kernel(
    const float* __restrict__ Yg, const float* __restrict__ Ug,
    const float* __restrict__ maskg, const float* __restrict__ Ag,
    const float* __restrict__ Bmg, const float* __restrict__ Cg,
    const float* __restrict__ mu0g, const float* __restrict__ Sig0g,
    const float* __restrict__ Qg, const float* __restrict__ Rg,
    float* __restrict__ out) {
  __shared__ float sA[32 * LD], sC[32 * LD], sQ[32 * LD], sRm[32 * LD];
  __shared__ float sSig[32 * LD], sSigP[32 * LD], sT[32 * LD], sT4[32 * LD];
  __shared__ float sS[32 * LD], sK[32 * LD], sKR[32 * LD], sNew[32 * LD];
  __shared__ float sBm[32 * 16];
  __shared__ float sMu[32], sMuP[32], sRv[32];

  const int tid = threadIdx.x;
  const int b = blockIdx.x;

  const size_t BT = (size_t)Bb * Tt;
  float* mf = out;                    // [B,T,32]
  float* Sf = out + BT * 32;          // [B,T,32,32]
  float* mp = Sf + BT * 1024;         // [B,T,32]
  float* Sp = mp + BT * 32;           // [B,T,32,32]

  // ---- init: constants + state into LDS ----
  for (int e = tid; e < 1024; e += 128) {
    int i = e >> 5, j = e & 31;
    sA[i * LD + j] = Ag[e];
    sC[i * LD + j] = Cg[e];
    sQ[i * LD + j] = Qg[e];
    sRm[i * LD + j] = Rg[e];
    sSig[i * LD + j] = Sig0g[e];
  }
  for (int e = tid; e < 512; e += 128) sBm[e] = Bmg[e];
  if (tid < 32) sMu[tid] = mu0g[tid];
  __syncthreads();

#pragma unroll 1
  for (int t = 0; t < Tt; ++t) {
    const size_t bt = (size_t)b * Tt + t;

    // S1: T = A*Sigma   | wave0 lanes: mu_pred = A*mu + Bm*u
    mm32(sT, sA, sSig, nullptr, false);
    if (tid < 32) {
      float acc = 0.f;
      const float* u = Ug + bt * Mm;
#pragma unroll
      for (int j = 0; j < 32; ++j) acc += sA[tid * LD + j] * sMu[j];
#pragma unroll
      for (int j = 0; j < 16; ++j) acc += sBm[tid * 16 + j] * u[j];
      sMuP[tid] = acc;
    }
    __syncthreads();

    // S2: Sigma_pred = T*A^T + Q (Q seeds the accumulator)
    mm32(sSigP, sT, sA, sQ, true);
    __syncthreads();

    // write predictions; r = y - C*mu_pred; S3: T = C*Sigma_pred
    {
      float* dst = Sp + bt * 1024;
      for (int e = tid; e < 1024; e += 128) dst[e] = sSigP[(e >> 5) * LD + (e & 31)];
      if (tid < 32) {
        mp[bt * 32 + tid] = sMuP[tid];
        float acc = Yg[bt * Pp + tid];
#pragma unroll
        for (int n = 0; n < 32; ++n) acc -= sC[tid * LD + n] * sMuP[n];
        sRv[tid] = acc;
      }
    }
    mm32(sT, sC, sSigP, nullptr, false);
    __syncthreads();

    // S4: S = T*C^T + R  |  S5: PCT = Sigma_pred*C^T  (into sK)
    mm32(sS, sT, sC, sRm, true);
    mm32(sK, sSigP, sC, nullptr, true);
    __syncthreads();

    symmetrize(sS);  // S = 0.5*(S + S^T)

    // Cholesky S = L L^T in-place (lower triangle). lane = row; column
    // broadcast via shuffles; one workgroup barrier per column.
#pragma unroll 1
    for (int k = 0; k < 32; ++k) {
      __syncthreads();
      if (tid < 32) {
        float d = sqrtf(sS[k * LD + k]);
        float Lik = 0.f;
        if (tid == k)
          Lik = d;
        else if (tid > k)
          Lik = sS[tid * LD + k] / d;
        if (tid >= k) sS[tid * LD + k] = Lik;
        for (int j = k + 1; j < 32; ++j) {
          float ljk = __shfl(Lik, j, 32);
          if (tid >= j) sS[tid * LD + j] -= Lik * ljk;
        }
      }
    }
    __syncthreads();

    // Triangular solves: K = PCT * S^-1 = ((PCT L^-T) L^-1), lane = row of K.
    // Row kept in 32 VGPRs (fully unrolled); L entries are LDS broadcasts.
    const float mval = maskg[bt];
    if (tid < 32) {
      float w[32];
#pragma unroll
      for (int j = 0; j < 32; ++j) w[j] = sK[tid * LD + j];
#pragma unroll
      for (int j = 0; j < 32; ++j) {  // W L^T = PCT
        float s = w[j];
#pragma unroll
        for (int m2 = 0; m2 < j; ++m2) s -= w[m2] * sS[j * LD + m2];
        w[j] = s / sS[j * LD + j];
      }
#pragma unroll
      for (int j = 31; j >= 0; --j) {  // K L = W
        float s = w[j];
#pragma unroll
        for (int m2 = j + 1; m2 < 32; ++m2) s -= w[m2] * sS[m2 * LD + j];
        w[j] = s / sS[j * LD + j];
      }
#pragma unroll
      for (int j = 0; j < 32; ++j) sK[tid * LD + j] = w[j] * mval;  // mask obs
    }
    __syncthreads();

    // mu_new = mu_pred + K*r ; S7: T = K*C
    if (tid < 32) {
      float acc = sMuP[tid];
#pragma unroll
      for (int p2 = 0; p2 < 32; ++p2) acc += sK[tid * LD + p2] * sRv[p2];
      sMu[tid] = acc;
      mf[bt * 32 + tid] = acc;
    }
    mm32(sT, sK, sC, nullptr, false);
    __syncthreads();

    // I - K*C in place on sT
    for (int e = tid; e < 1024; e += 128) {
      int i = e >> 5, j = e & 31;
      float v = -sT[i * LD + j];
      if (i == j) v += 1.f;
      sT[i * LD + j] = v;
    }
    __syncthreads();

    // S8: T4 = (I-KC)*Sigma_pred  |  S10: KR = K*R
    mm32(sT4, sT, sSigP, nullptr, false);
    mm32(sKR, sK, sRm, nullptr, false);
    __syncthreads();

    // S9+S11 fused: Sigma_new = T4*(I-KC)^T + KR*K^T
    mm32_two_bt(sNew, sT4, sT, sKR, sK);
    __syncthreads();

    symmetrize(sNew);
    __syncthreads();

    // commit filtered covariance + write outputs
    {
      float* dst = Sf + bt * 1024;
      for (int e = tid; e < 1024; e += 128) {
        int i = e >> 5, j = e & 31;
        float v = sNew[i * LD + j];
        sSig[i * LD + j] = v;
        dst[e] = v;
      }
    }
    __syncthreads();
  }
}

extern "C" void kernel_launch(void* const* d_in, const int* in_sizes, int n_in,
                              void* d_out, int out_size, void* d_ws, size_t ws_size,
                              hipStream_t stream) {
  (void)in_sizes;
  (void)n_in;
  (void)out_size;
  (void)d_ws;
  (void)ws_size;
  const float* Yg = (const float*)d_in[0];
  const float* Ug = (const float*)d_in[1];
  const float* maskg = (const float*)d_in[2];
  const float* Ag = (const float*)d_in[3];
  const float* Bmg = (const float*)d_in[4];
  const float* Cg = (const float*)d_in[5];
  const float* mu0g = (const float*)d_in[6];
  const float* Sig0g = (const float*)d_in[7];
  const float* Qg = (const float*)d_in[8];
  const float* Rg = (const float*)d_in[9];
  kalman_fwd_kernel<<<dim3(Bb), dim3(128), 0, stream>>>(
      Yg, Ug, maskg, Ag, Bmg, Cg, mu0g, Sig0g, Qg, Rg, (float*)d_out);
}